// YOLOv10PostProcess_61821759258643
// MI455X (gfx1250) — compile-verified
//
#include <hip/hip_runtime.h>
#include <stdint.h>

#define NBOX   33600
#define NBAT   128
#define NCH    6
#define KTOP   300
#define NTHR   1024
#define CHUNK  33          // ceil(33600 / 1024)
#define INVSZ  (1.0f / 1280.0f)

// Monotonic float->uint mapping: unsigned compare order == float compare order.
__device__ __forceinline__ uint32_t fkey(uint32_t u) {
    return u ^ (uint32_t)(((int32_t)u >> 31) | (int32_t)0x80000000);
}

__global__ void __launch_bounds__(NTHR, 1)
yolo_topk_post(const float* __restrict__ preds, float* __restrict__ out) {
    __shared__ uint32_t s_scores[NBOX];       // raw fp32 bits of channel 4
    __shared__ uint32_t s_hist[256];
    __shared__ uint32_t s_scan[NTHR];
    __shared__ uint64_t s_sort[512];
    __shared__ uint32_t s_prefix, s_rem, s_cnt;

    const int tid = threadIdx.x;
    const int b   = blockIdx.x;
    const float* __restrict__ row = preds + (size_t)b * NBOX * NCH;

    // ---- Phase A: async strided copy of score channel into LDS (CDNA5 path) ----
    {
        uint32_t lds_base = (uint32_t)(uintptr_t)(&s_scores[0]); // low 32b of flat LDS ptr = LDS offset
        for (int i = tid; i < NBOX; i += NTHR) {
            uint32_t ldsAddr = lds_base + 4u * (uint32_t)i;
            uint32_t gOff    = 24u * (uint32_t)i + 16u;          // byte offset of preds[b][i][4]
            asm volatile("global_load_async_to_lds_b32 %0, %1, %2"
                         :: "v"(ldsAddr), "v"(gOff), "s"(row) : "memory");
        }
        asm volatile("s_wait_asynccnt 0" ::: "memory");
    }
    __syncthreads();

    const int begin = tid * CHUNK;
    const int end   = (begin + CHUNK < NBOX) ? (begin + CHUNK) : NBOX;

    // ---- Phase B: radix-select the 300th-largest key (exact, deterministic) ----
    uint32_t prefix    = 0;
    uint32_t remaining = KTOP;
    for (int p = 0; p < 4; ++p) {
        const int shift = 24 - 8 * p;
        if (tid < 256) s_hist[tid] = 0u;
        __syncthreads();
        for (int i = begin; i < end; ++i) {
            uint32_t key = fkey(s_scores[i]);
            if (p == 0 || (key >> (shift + 8)) == prefix) {
                atomicAdd(&s_hist[(key >> shift) & 255u], 1u);
            }
        }
        __syncthreads();
        if (tid == 0) {
            uint32_t cum = 0;
            for (int bin = 255; bin >= 0; --bin) {
                uint32_t c = s_hist[bin];
                if (cum + c >= remaining) {
                    s_rem    = remaining - cum;
                    s_prefix = (prefix << 8) | (uint32_t)bin;
                    break;
                }
                cum += c;
            }
        }
        __syncthreads();
        prefix    = s_prefix;
        remaining = s_rem;
        __syncthreads();
    }
    const uint32_t T    = prefix;          // exact key of the K-th largest
    const uint32_t need = remaining;       // # ties at T to take (lowest indices first)
    const uint32_t G    = KTOP - need;     // # strictly greater than T

    // ---- Phase C: collect winners into 64-bit items (key<<32 | ~index) ----
    if (tid == 0) s_cnt = 0u;
    if (tid < 512) s_sort[tid] = 0ull;     // pad items sort last
    __syncthreads();

    for (int i = begin; i < end; ++i) {
        uint32_t key = fkey(s_scores[i]);
        if (key > T) {
            uint32_t pos = atomicAdd(&s_cnt, 1u);
            s_sort[pos] = ((uint64_t)key << 32) | (uint32_t)(~(uint32_t)i);
        }
    }
    // Tie selection: contiguous per-thread ranges -> global index order via prefix scan
    uint32_t myties = 0;
    for (int i = begin; i < end; ++i) myties += (fkey(s_scores[i]) == T) ? 1u : 0u;
    s_scan[tid] = myties;
    __syncthreads();
    for (int off = 1; off < NTHR; off <<= 1) {
        uint32_t v = s_scan[tid];
        uint32_t w = (tid >= off) ? s_scan[tid - off] : 0u;
        __syncthreads();
        s_scan[tid] = v + w;
        __syncthreads();
    }
    uint32_t tieBase = s_scan[tid] - myties;   // exclusive rank of my first tie
    uint32_t lr = 0;
    for (int i = begin; i < end; ++i) {
        if (fkey(s_scores[i]) == T) {
            uint32_t g = tieBase + lr;
            if (g < need) s_sort[G + g] = ((uint64_t)T << 32) | (uint32_t)(~(uint32_t)i);
            ++lr;
        }
    }
    __syncthreads();

    // ---- Phase D: bitonic sort 512 items descending (key desc, index asc) ----
    for (int k = 2; k <= 512; k <<= 1) {
        for (int j = k >> 1; j > 0; j >>= 1) {
            __syncthreads();
            if (tid < 256) {
                int i   = ((tid & ~(j - 1)) << 1) | (tid & (j - 1));
                int ixj = i | j;
                uint64_t a = s_sort[i];
                uint64_t c = s_sort[ixj];
                bool up = ((i & k) == 0);              // descending network
                if (up ? (a < c) : (a > c)) { s_sort[i] = c; s_sort[ixj] = a; }
            }
        }
    }
    __syncthreads();

    // ---- Phase E: gather (L2-resident) + transform + store ----
    if (tid < KTOP) {
        uint64_t item = s_sort[tid];
        uint32_t idx  = ~((uint32_t)item);
        const float* __restrict__ p = row + (size_t)idx * NCH;
        float x1 = p[0] * INVSZ;
        float y1 = p[1] * INVSZ;
        float x2 = p[2] * INVSZ;
        float y2 = p[3] * INVSZ;
        float conf = p[4];
        float cls  = p[5];
        float* __restrict__ o = out + ((size_t)b * KTOP + (size_t)tid) * NCH;
        o[0] = (x1 + x2) * 0.5f;
        o[1] = (y1 + y2) * 0.5f;
        o[2] = x2 - x1;
        o[3] = y2 - y1;
        o[4] = conf;
        o[5] = cls;
    }
}

extern "C" void kernel_launch(void* const* d_in, const int* in_sizes, int n_in,
                              void* d_out, int out_size, void* d_ws, size_t ws_size,
                              hipStream_t stream) {
    (void)in_sizes; (void)n_in; (void)d_ws; (void)ws_size; (void)out_size;
    const float* preds = (const float*)d_in[0];
    float* out = (float*)d_out;
    yolo_topk_post<<<dim3(NBAT), dim3(NTHR), 0, stream>>>(preds, out);
}